// HCCFEncoder_12429635354857
// MI455X (gfx1250) — compile-verified
//
#include <hip/hip_runtime.h>

// ---- problem constants (from reference) ----
#define N_USERS   100000
#define N_ITEMS   200000
#define N_NODES   300000
#define DD        64
#define HYPER     128
#define NNZ_      4000000
#define NODE_ELEMS (N_NODES * DD)       // 19,200,000 floats
#define UELEMS     (N_USERS * DD)       // 6,400,000

typedef float v2f __attribute__((ext_vector_type(2)));
typedef float v8f __attribute__((ext_vector_type(8)));

// ---------------------------------------------------------------------------
// Zero a float buffer (grid-stride).
__global__ void k_zero(float* __restrict__ p, long n) {
  long i = (long)blockIdx.x * blockDim.x + threadIdx.x;
  long stride = (long)gridDim.x * blockDim.x;
  for (; i < n; i += stride) p[i] = 0.0f;
}

// ---------------------------------------------------------------------------
// h = concat(user_emb, item_emb); hidden_sum = h.  (float4; boundary at UELEMS
// is float4-aligned)
__global__ void k_init(const float* __restrict__ ue, const float* __restrict__ ie,
                       float* __restrict__ h, float* __restrict__ sum) {
  long i4 = (long)blockIdx.x * blockDim.x + threadIdx.x;
  long stride = (long)gridDim.x * blockDim.x;
  long n4 = NODE_ELEMS / 4;
  for (; i4 < n4; i4 += stride) {
    long i = i4 * 4;
    float4 v;
    if (i < UELEMS) v = *(const float4*)(ue + i);
    else            v = *(const float4*)(ie + (i - UELEMS));
    *(float4*)(h + i)   = v;
    *(float4*)(sum + i) = v;
  }
}

// ---------------------------------------------------------------------------
// Row-major fp32 GEMM via V_WMMA_F32_16X16X4_F32.
// C[M x N] = A[M x K] * B[K x N].  N, K compile-time (N in {64,128}, N*K=8192).
//
// B is staged in LDS (32 KB) once per block, pre-swizzled into K-pair
// interleaved layout:  Bs[(k/2)*2N + 2n + (k&1)] = B[k][n]
// so a WMMA B operand {B[k][n], B[k+1][n]} is ONE aligned ds_load_b64.
//
// Each wave computes a 16x64 strip (4 N-tiles, 4 accumulators): one A float2
// load feeds 4 WMMAs.  Block = 256 threads = 8 waves sharing the LDS B.
template <int N, int K>
__global__ void k_gemm_wmma(const float* __restrict__ A, const float* __restrict__ B,
                            float* __restrict__ C, int M) {
  __shared__ float Bs[N * K];            // 8192 floats = 32 KB
  for (int i = threadIdx.x; i < N * K; i += blockDim.x) {
    int k = i / N, n = i - k * N;
    Bs[(k >> 1) * (2 * N) + (n << 1) + (k & 1)] = B[i];
  }
  __syncthreads();

  constexpr int CG = N / 64;             // column groups of 64
  int wave = (int)((blockIdx.x * blockDim.x + threadIdx.x) >> 5);
  int lane = threadIdx.x & 31;
  int mtiles = M >> 4;
  int mt = wave / CG;
  int cg = wave - mt * CG;
  if (mt >= mtiles) return;              // wave-uniform: EXEC stays all-ones

  int m  = lane & 15;                    // row within tile (A), col within tile (B/C)
  int kk = (lane >> 4) << 1;             // lane-half K offset: 0 or 2

  const float* Arow = A + (long)(mt * 16 + m) * K;
  int nbase = cg * 64 + m;

  v8f c0 = {}, c1 = {}, c2 = {}, c3 = {};
  for (int k0 = 0; k0 < K; k0 += 4) {
    v2f a = *(const v2f*)(Arow + k0 + kk);                 // global_load_b64
    // swizzled B: pair (k0+kk, k0+kk+1) at float index (k0+kk)*N + 2n
    const float* bp = &Bs[(k0 + kk) * N + (nbase << 1)];
    v2f b;
    b = *(const v2f*)(bp +  0);                            // ds_load_b64
    c0 = __builtin_amdgcn_wmma_f32_16x16x4_f32(false, a, false, b, (short)0, c0, false, false);
    b = *(const v2f*)(bp + 32);
    c1 = __builtin_amdgcn_wmma_f32_16x16x4_f32(false, a, false, b, (short)0, c1, false, false);
    b = *(const v2f*)(bp + 64);
    c2 = __builtin_amdgcn_wmma_f32_16x16x4_f32(false, a, false, b, (short)0, c2, false, false);
    b = *(const v2f*)(bp + 96);
    c3 = __builtin_amdgcn_wmma_f32_16x16x4_f32(false, a, false, b, (short)0, c3, false, false);
  }

  // C/D layout per tile: VGPR r -> row mt*16 + r + 8*(lane/16), col = tile_col + lane%16
  int rbase = mt * 16 + ((lane >> 4) << 3);
#pragma unroll
  for (int r = 0; r < 8; ++r) {
    long ro = (long)(rbase + r) * N + nbase;
    C[ro +  0] = c0[r];
    C[ro + 16] = c1[r];
    C[ro + 32] = c2[r];
    C[ro + 48] = c3[r];
  }
}

// ---------------------------------------------------------------------------
// SpMM scatter: gcn[row,:] += val * h[col,:].  8 threads per nonzero,
// each thread handles 8 contiguous of the 64 columns (2x float4 gather,
// 8x global_atomic_add_f32).
__global__ void k_spmm(const float* __restrict__ vals, const int* __restrict__ rows,
                       const int* __restrict__ cols, const float* __restrict__ h,
                       float* __restrict__ gcn) {
  int e   = blockIdx.x * 32 + (threadIdx.x >> 3);
  int sub = threadIdx.x & 7;
  if (e >= NNZ_) return;
  float v  = vals[e];
  int  col = cols[e];
  int  row = rows[e];
  const float4* hp = (const float4*)(h + (long)col * DD + sub * 8);
  float4 x0 = hp[0], x1 = hp[1];
  float* gp = gcn + (long)row * DD + sub * 8;
  unsafeAtomicAdd(gp + 0, v * x0.x);
  unsafeAtomicAdd(gp + 1, v * x0.y);
  unsafeAtomicAdd(gp + 2, v * x0.z);
  unsafeAtomicAdd(gp + 3, v * x0.w);
  unsafeAtomicAdd(gp + 4, v * x1.x);
  unsafeAtomicAdd(gp + 5, v * x1.y);
  unsafeAtomicAdd(gp + 6, v * x1.z);
  unsafeAtomicAdd(gp + 7, v * x1.w);
}

// ---------------------------------------------------------------------------
// S[128 x 64] += Hyp[Nrows x 128]^T @ X[Nrows x 64].
// Block = 256 threads processes CHUNK rows: X chunk staged in LDS,
// thread owns (hyperdim = tid%128, 32 of the 64 cols); atomic reduce at end.
#define CHUNK 192
__global__ void k_atx(const float* __restrict__ Hyp, const float* __restrict__ X,
                      float* __restrict__ S, int Nrows) {
  __shared__ float lds[CHUNK * DD];      // 48 KB
  int base = blockIdx.x * CHUNK;
  int cnt  = Nrows - base; if (cnt > CHUNK) cnt = CHUNK;

  for (int i = threadIdx.x; i < cnt * DD; i += 256)
    lds[i] = X[(long)base * DD + i];
  __syncthreads();

  int hd = threadIdx.x & 127;
  int d0 = (threadIdx.x >> 7) * 32;

  float acc[32];
#pragma unroll
  for (int j = 0; j < 32; ++j) acc[j] = 0.0f;

  for (int k = 0; k < cnt; ++k) {
    float a = Hyp[(long)(base + k) * HYPER + hd];
    const float* xr = &lds[k * DD + d0];
#pragma unroll
    for (int j = 0; j < 32; ++j) acc[j] += a * xr[j];
  }

#pragma unroll
  for (int j = 0; j < 32; ++j)
    unsafeAtomicAdd(&S[hd * DD + d0 + j], acc[j]);
}

// ---------------------------------------------------------------------------
// h = gcn + hg; hidden_sum += h.   (float4 grid-stride)
__global__ void k_fuse(const float* __restrict__ gcn, const float* __restrict__ hg,
                       float* __restrict__ h, float* __restrict__ sum) {
  long i4 = (long)blockIdx.x * blockDim.x + threadIdx.x;
  long stride = (long)gridDim.x * blockDim.x;
  long n4 = NODE_ELEMS / 4;
  for (; i4 < n4; i4 += stride) {
    long i = i4 * 4;
    float4 g = *(const float4*)(gcn + i);
    float4 q = *(const float4*)(hg + i);
    float4 v = make_float4(g.x + q.x, g.y + q.y, g.z + q.z, g.w + q.w);
    *(float4*)(h + i) = v;
    float4 s = *(float4*)(sum + i);
    s.x += v.x; s.y += v.y; s.z += v.z; s.w += v.w;
    *(float4*)(sum + i) = s;
  }
}

// ---------------------------------------------------------------------------
extern "C" void kernel_launch(void* const* d_in, const int* in_sizes, int n_in,
                              void* d_out, int out_size, void* d_ws, size_t ws_size,
                              hipStream_t stream) {
  const float* user_emb = (const float*)d_in[0];
  const float* item_emb = (const float*)d_in[1];
  const float* user_w   = (const float*)d_in[2];
  const float* item_w   = (const float*)d_in[3];
  const float* adj_vals = (const float*)d_in[4];
  const int*   adj_rows = (const int*)d_in[5];
  const int*   adj_cols = (const int*)d_in[6];

  float* out = (float*)d_out;
  float* sum      = out;                               // [:300000,64]  (user_out|item_out)
  float* gcn_base = out + (long)NODE_ELEMS;            // gcn_hidden[2]
  float* hg_base  = out + (long)NODE_ELEMS * 3;        // hgnn_hidden[2]

  float* ws = (float*)d_ws;
  float* h        = ws;                                // 19.2M
  float* hyper_uu = h + (long)NODE_ELEMS;              // 12.8M
  float* hyper_ii = hyper_uu + (long)N_USERS * HYPER;  // 25.6M
  float* small_u  = hyper_ii + (long)N_ITEMS * HYPER;  // 8192
  float* small_i  = small_u + HYPER * DD;              // 8192 (contiguous with small_u)

  const int gs = 4096;

  // h0 = concat(emb); hidden_sum = h0
  k_init<<<gs, 256, 0, stream>>>(user_emb, item_emb, h, sum);

  // hyper projections (WMMA f32): emb @ W   (N=128, K=64; 2 waves per 16-row strip)
  {
    int waves_u = (N_USERS / 16) * 2;
    k_gemm_wmma<HYPER, DD><<<(waves_u + 7) / 8, 256, 0, stream>>>(user_emb, user_w,
                                                                  hyper_uu, N_USERS);
    int waves_i = (N_ITEMS / 16) * 2;
    k_gemm_wmma<HYPER, DD><<<(waves_i + 7) / 8, 256, 0, stream>>>(item_emb, item_w,
                                                                  hyper_ii, N_ITEMS);
  }

  for (int l = 0; l < 2; ++l) {
    float* gcn = gcn_base + (long)l * NODE_ELEMS;
    float* hg  = hg_base  + (long)l * NODE_ELEMS;

    // GCN branch: gcn = segment_sum(adj_vals * h[cols], rows)
    k_zero<<<gs, 256, 0, stream>>>(gcn, (long)NODE_ELEMS);
    k_spmm<<<NNZ_ / 32, 256, 0, stream>>>(adj_vals, adj_rows, adj_cols, h, gcn);

    // Hyper branch: small = Hyp^T @ h_part  (atomic FMA reduction)
    k_zero<<<64, 256, 0, stream>>>(small_u, (long)(HYPER * DD * 2));
    k_atx<<<(N_USERS + CHUNK - 1) / CHUNK, 256, 0, stream>>>(hyper_uu, h, small_u, N_USERS);
    k_atx<<<(N_ITEMS + CHUNK - 1) / CHUNK, 256, 0, stream>>>(hyper_ii, h + (long)UELEMS,
                                                             small_i, N_ITEMS);

    // hg = Hyp @ small  (WMMA f32 expansion; N=64, K=128; 1 wave per 16-row strip)
    {
      int waves_u = N_USERS / 16;
      k_gemm_wmma<DD, HYPER><<<(waves_u + 7) / 8, 256, 0, stream>>>(hyper_uu, small_u,
                                                                    hg, N_USERS);
      int waves_i = N_ITEMS / 16;
      k_gemm_wmma<DD, HYPER><<<(waves_i + 7) / 8, 256, 0, stream>>>(hyper_ii, small_i,
                                                                    hg + (long)UELEMS,
                                                                    N_ITEMS);
    }

    // h = gcn + hg; hidden_sum += h
    k_fuse<<<gs, 256, 0, stream>>>(gcn, hg, h, sum);
  }
}